// FeatureInteraction_5403068858956
// MI455X (gfx1250) — compile-verified
//
#include <hip/hip_runtime.h>

typedef __attribute__((ext_vector_type(2))) float v2f;
typedef __attribute__((ext_vector_type(4))) float v4f;
typedef __attribute__((ext_vector_type(8))) float v8f;

#define NF 27           // features: 1 bottom + 26 embeddings
#define ND 128          // embedding dim
#define NPAIRS 351      // 27*26/2
#define ROWSTRIDE 132   // 128 + 4 floats pad -> bank-conflict-free ds_load_b64
#define ROWS_PAD 28     // rows 0..26 data, row 27 zeros (pad source for tile1)
#define WAVES_PER_BLOCK 4

__device__ __forceinline__ int pair_index(int i, int j) {
  // p = i*26 - i*(i-1)/2 + (j-i-1) == i*(2*NF-i-1)/2 + (j-i-1)
  return i * (2 * NF - i - 1) / 2 + (j - i - 1);
}

__global__ __launch_bounds__(WAVES_PER_BLOCK * 32)
void gram_triu_kernel(const float* __restrict__ bottom,
                      const float* __restrict__ emb,
                      float* __restrict__ out, int nbatch) {
  __shared__ float lds[WAVES_PER_BLOCK * ROWS_PAD * ROWSTRIDE];  // 59136 B

  const int wave = threadIdx.x >> 5;
  const int lane = threadIdx.x & 31;
  const int b = blockIdx.x * WAVES_PER_BLOCK + wave;  // wave-uniform guard, EXEC stays full
  if (b >= nbatch) return;

  float* my = &lds[wave * ROWS_PAD * ROWSTRIDE];
  const float* sb = bottom + (size_t)b * ND;
  const float* se = emb + (size_t)b * 26 * ND;

  // ---- Stage batch into LDS: coalesced 128-bit streaming loads (non-temporal:
  // 226 MB input > 192 MB L2, touched once). 512 B contiguous per row per wave.
  {
    v4f v0 = __builtin_nontemporal_load((const v4f*)(sb + 4 * lane));
    *(v4f*)(my + 4 * lane) = v0;
#pragma unroll
    for (int r = 1; r < NF; ++r) {
      v4f e = __builtin_nontemporal_load(
          (const v4f*)(se + (size_t)(r - 1) * ND + 4 * lane));
      *(v4f*)(my + r * ROWSTRIDE + 4 * lane) = e;
    }
    v4f z = {0.f, 0.f, 0.f, 0.f};
    *(v4f*)(my + 27 * ROWSTRIDE + 4 * lane) = z;  // zero pad row
  }
  // No barrier: wave-private LDS region; LDS ops are in-order within a wave.

  // ---- A-layout fragments for V_WMMA_F32_16X16X4_F32:
  // lane<16  : row M=lane,    VGPR0/1 = K,K+1
  // lane>=16 : row M=lane-16, VGPR0/1 = K+2,K+3
  // B operand layout is the same structure with N in place of M, so passing a
  // row-fragment as both A and B computes R * R^T directly.
  const int half = lane >> 4;
  const int row = lane & 15;
  const float* p0 = my + row * ROWSTRIDE + 2 * half;             // tile0: rows 0..15
  const int r1 = (16 + row > 27) ? 27 : (16 + row);              // tile1: rows 16..26, pad->27 (zeros)
  const float* p1 = my + r1 * ROWSTRIDE + 2 * half;

  v8f c00 = {};  // gram[0:16, 0:16]
  v8f c01 = {};  // gram[0:16, 16:32]
  v8f c11 = {};  // gram[16:32, 16:32]
#pragma unroll
  for (int k = 0; k < ND; k += 4) {
    v2f a0 = *(const v2f*)(p0 + k);
    v2f a1 = *(const v2f*)(p1 + k);
    c00 = __builtin_amdgcn_wmma_f32_16x16x4_f32(false, a0, false, a0,
                                                (short)0, c00, false, false);
    c01 = __builtin_amdgcn_wmma_f32_16x16x4_f32(false, a0, false, a1,
                                                (short)0, c01, false, false);
    c11 = __builtin_amdgcn_wmma_f32_16x16x4_f32(false, a1, false, a1,
                                                (short)0, c11, false, false);
  }

  // ---- Scatter strict upper triangle. C/D layout: element v of lane L holds
  // M = v + 8*(L>=16), N = L%16.
  float* ob = out + (size_t)b * NPAIRS;
  const int col = lane & 15;
#pragma unroll
  for (int v = 0; v < 8; ++v) {
    const int m = v + 8 * half;
    if (col > m)
      __builtin_nontemporal_store(c00[v], ob + pair_index(m, col));
    const int j1 = 16 + col;
    if (j1 < NF)
      __builtin_nontemporal_store(c01[v], ob + pair_index(m, j1));
    const int i1 = 16 + m;
    if (i1 < NF && j1 < NF && col > m)
      __builtin_nontemporal_store(c11[v], ob + pair_index(i1, j1));
  }
}

extern "C" void kernel_launch(void* const* d_in, const int* in_sizes, int n_in,
                              void* d_out, int out_size, void* d_ws, size_t ws_size,
                              hipStream_t stream) {
  const float* bottom = (const float*)d_in[0];  // (B, 128) fp32
  const float* emb    = (const float*)d_in[1];  // (B, 26, 128) fp32
  float* out          = (float*)d_out;          // (B, 351) fp32

  const int nbatch = in_sizes[0] / ND;  // 16384
  const int blocks = (nbatch + WAVES_PER_BLOCK - 1) / WAVES_PER_BLOCK;
  gram_triu_kernel<<<blocks, WAVES_PER_BLOCK * 32, 0, stream>>>(bottom, emb, out, nbatch);
}